// Decoder_8435315769624
// MI455X (gfx1250) — compile-verified
//
#include <hip/hip_runtime.h>
#include <hip/hip_bf16.h>
#include <math.h>

typedef __attribute__((ext_vector_type(2))) float v2f;
typedef __attribute__((ext_vector_type(8))) float v8f;

// ---------------------------------------------------------------------------
// WMMA fp32 GEMM:  C[M,N] = A[M,K] @ W[N,K]^T + bias[N]   (optional ReLU)
// A row-major [M,K], W row-major [N,K] (torch weight layout), C row-major [M,N].
//
// Block = 256 threads (8 waves), tile 128(M) x 64(N), K-panel = 32.
// Panels are staged in LDS (double buffered): global loads for panel p+1 are
// issued before computing panel p, so the loadcnt wait is covered by 32 WMMAs
// of compute. Fragments are read from LDS (ds_load_b64), and all 8 waves share
// one W panel. LDS row stride 36 floats => 8/16B-aligned accesses and no bank
// conflicts (36*i mod 64 distinct for i=0..15; +2-dword offset never collides).
// M must be a multiple of 128 (call sites: 30720, 512). K-tail (K=300) is
// zero-padded in LDS so the inner loop has no guards (EXEC all-ones for WMMA).
// ---------------------------------------------------------------------------
#define BM 128
#define BN 64
#define BK 32
#define LDSS 36   // padded row stride (floats)

__global__ __launch_bounds__(256) void gemm_wmma_f32(
    const float* __restrict__ A, const float* __restrict__ W,
    const float* __restrict__ bias, float* __restrict__ C,
    int M, int N, int K, int relu)
{
  __shared__ float As[2][BM * LDSS];
  __shared__ float Ws[2][BN * LDSS];

  const int tid   = threadIdx.x;
  const int lane  = tid & 31;
  const int wave  = tid >> 5;
  const int m0    = blockIdx.y * BM;
  const int n0    = blockIdx.x * BN;
  const int khalf = lane >> 4;            // 0: K={0,1}, 1: K={2,3} within step
  const int l16   = lane & 15;

  v8f acc[4] = {};

  const int nPanels = (K + BK - 1) / BK;

  // register staging for the next panel: A = 1024 float4 (4/thread),
  // W = 512 float4 (2/thread)
  float4 ga[4], gw[2];

  auto loadPanel = [&](int p) {
    const int k0 = p * BK;
    #pragma unroll
    for (int i = 0; i < 4; ++i) {
      const int idx = tid + i * 256;      // [0, 1024)
      const int row = idx >> 3;           // 0..127
      const int gk  = k0 + (idx & 7) * 4; // col in K
      const int gm  = m0 + row;
      if (gm < M && gk + 4 <= K) {
        ga[i] = *(const float4*)(A + (size_t)gm * K + gk);
      } else {
        float4 v;
        v.x = (gm < M && gk + 0 < K) ? A[(size_t)gm * K + gk + 0] : 0.0f;
        v.y = (gm < M && gk + 1 < K) ? A[(size_t)gm * K + gk + 1] : 0.0f;
        v.z = (gm < M && gk + 2 < K) ? A[(size_t)gm * K + gk + 2] : 0.0f;
        v.w = (gm < M && gk + 3 < K) ? A[(size_t)gm * K + gk + 3] : 0.0f;
        ga[i] = v;
      }
    }
    #pragma unroll
    for (int i = 0; i < 2; ++i) {
      const int idx = tid + i * 256;      // [0, 512)
      const int row = idx >> 3;           // 0..63
      const int gk  = k0 + (idx & 7) * 4;
      const int gn  = n0 + row;
      if (gn < N && gk + 4 <= K) {
        gw[i] = *(const float4*)(W + (size_t)gn * K + gk);
      } else {
        float4 v;
        v.x = (gn < N && gk + 0 < K) ? W[(size_t)gn * K + gk + 0] : 0.0f;
        v.y = (gn < N && gk + 1 < K) ? W[(size_t)gn * K + gk + 1] : 0.0f;
        v.z = (gn < N && gk + 2 < K) ? W[(size_t)gn * K + gk + 2] : 0.0f;
        v.w = (gn < N && gk + 3 < K) ? W[(size_t)gn * K + gk + 3] : 0.0f;
        gw[i] = v;
      }
    }
  };

  auto stashPanel = [&](int buf) {
    #pragma unroll
    for (int i = 0; i < 4; ++i) {
      const int idx = tid + i * 256;
      const int row = idx >> 3;
      const int c4  = (idx & 7) * 4;
      *(float4*)&As[buf][row * LDSS + c4] = ga[i];   // 16B aligned (36*4=144)
    }
    #pragma unroll
    for (int i = 0; i < 2; ++i) {
      const int idx = tid + i * 256;
      const int row = idx >> 3;
      const int c4  = (idx & 7) * 4;
      *(float4*)&Ws[buf][row * LDSS + c4] = gw[i];
    }
  };

  // prime panel 0
  loadPanel(0);
  stashPanel(0);
  __syncthreads();

  const float* Abase = &As[0][(wave * 16 + l16) * LDSS];
  const float* Wb[4];
  #pragma unroll
  for (int j = 0; j < 4; ++j) Wb[j] = &Ws[0][(j * 16 + l16) * LDSS];
  const size_t bufStrideA = (size_t)BM * LDSS;
  const size_t bufStrideW = (size_t)BN * LDSS;

  for (int p = 0; p < nPanels; ++p) {
    const int cur = p & 1;

    if (p + 1 < nPanels) loadPanel(p + 1);          // global loads in flight
    if (p + 2 < nPanels)                            // pre-warm L2 two panels out
      __builtin_prefetch(A + (size_t)(m0 + (tid >> 3)) * K + (p + 2) * BK, 0, 3);

    const float* Ab = Abase + cur * bufStrideA;
    #pragma unroll
    for (int kk = 0; kk < BK; kk += 4) {
      const int k2 = kk + 2 * khalf;                // 8B-aligned LDS offset
      v2f a = *(const v2f*)(Ab + k2);
      #pragma unroll
      for (int j = 0; j < 4; ++j) {
        v2f b = *(const v2f*)(Wb[j] + cur * bufStrideW + k2);
        acc[j] = __builtin_amdgcn_wmma_f32_16x16x4_f32(
            /*neg_a=*/false, a, /*neg_b=*/false, b,
            /*c_mod=*/(short)0, acc[j], /*reuse_a=*/false, /*reuse_b=*/false);
      }
    }

    if (p + 1 < nPanels) stashPanel((p + 1) & 1);   // write other buffer
    __syncthreads();
  }

  // store (C/D layout: VGPR r -> M=r (lanes 0-15) / M=r+8 (lanes 16-31))
  const int mw = m0 + wave * 16;
  #pragma unroll
  for (int j = 0; j < 4; ++j) {
    const int n = n0 + j * 16 + l16;
    if (n >= N) continue;
    const float bv = bias ? bias[n] : 0.0f;
    #pragma unroll
    for (int r = 0; r < 8; ++r) {
      const int m = mw + r + 8 * khalf;
      if (m < M) {
        float v = acc[j][r] + bv;
        if (relu) v = fmaxf(v, 0.0f);
        C[(size_t)m * N + n] = v;
      }
    }
  }
}

// ---------------------------------------------------------------------------
// Attention: one block per (b, h, q). qkv rows are [3*Dm] = [Q | K | V],
// head h occupies columns [h*hd, (h+1)*hd) of each third.
// causal==1 -> mask keys k > q; causal==0 -> key-padding mask (encmask!=0).
// Output written head-concatenated: out[(b*L+q)*Dm + h*hd + d].
// ---------------------------------------------------------------------------
__global__ __launch_bounds__(64) void attn_kernel(
    const float* __restrict__ qkv, float* __restrict__ out,
    const int* __restrict__ encmask,
    int B, int H, int L, int hd, int Dm, int causal)
{
  const int bid = blockIdx.x;
  const int q = bid % L;
  const int h = (bid / L) % H;
  const int b = bid / (L * H);
  const int t = threadIdx.x;
  const int row3 = 3 * Dm;

  __shared__ float s[64];
  __shared__ float a[64];

  const float* Q = qkv + ((size_t)(b * L + q)) * row3 + h * hd;
  const float scale = rsqrtf((float)hd);

  if (t < L) {
    bool masked = causal ? (t > q) : (encmask[b * L + t] != 0);
    float acc = 0.0f;
    const float* Krow = qkv + ((size_t)(b * L + t)) * row3 + Dm + h * hd;
    for (int d = 0; d < hd; ++d) acc += Q[d] * Krow[d];
    s[t] = masked ? -1e30f : acc * scale;
  } else {
    s[t] = -1e30f;
  }
  __syncthreads();

  if (t == 0) {
    float mx = -1e30f;
    for (int k = 0; k < L; ++k) mx = fmaxf(mx, s[k]);
    float sum = 0.0f;
    for (int k = 0; k < L; ++k) { float e = expf(s[k] - mx); a[k] = e; sum += e; }
    float inv = 1.0f / sum;
    for (int k = 0; k < L; ++k) a[k] *= inv;
  }
  __syncthreads();

  float* Orow = out + ((size_t)(b * L + q)) * Dm + h * hd;
  for (int d = t; d < hd; d += 64) {
    float acc = 0.0f;
    for (int k = 0; k < L; ++k)
      acc += a[k] * qkv[((size_t)(b * L + k)) * row3 + 2 * Dm + h * hd + d];
    Orow[d] = acc;
  }
}

// ---------------------------------------------------------------------------
// Fused residual + LayerNorm over D=512.
// out = LN(x + pre) * g + beta + post     (pre/post may be null)
// One block (256 threads) per row; 2 elements per thread.
// ---------------------------------------------------------------------------
__global__ __launch_bounds__(256) void ln_kernel(
    float* __restrict__ out, const float* __restrict__ x,
    const float* __restrict__ pre, const float* __restrict__ post,
    const float* __restrict__ g, const float* __restrict__ beta)
{
  const int D = 512;
  const size_t row = (size_t)blockIdx.x * D;
  const int t = threadIdx.x;
  __shared__ float red[256];

  float v0 = x[row + t]       + (pre ? pre[row + t]       : 0.0f);
  float v1 = x[row + t + 256] + (pre ? pre[row + t + 256] : 0.0f);

  red[t] = v0 + v1;
  __syncthreads();
  for (int s2 = 128; s2 > 0; s2 >>= 1) { if (t < s2) red[t] += red[t + s2]; __syncthreads(); }
  float mu = red[0] * (1.0f / D);
  __syncthreads();

  float d0 = v0 - mu, d1 = v1 - mu;
  red[t] = d0 * d0 + d1 * d1;
  __syncthreads();
  for (int s2 = 128; s2 > 0; s2 >>= 1) { if (t < s2) red[t] += red[t + s2]; __syncthreads(); }
  float rstd = rsqrtf(red[0] * (1.0f / D) + 1e-5f);
  __syncthreads();

  out[row + t]       = d0 * rstd * g[t]       + beta[t]       + (post ? post[row + t]       : 0.0f);
  out[row + t + 256] = d1 * rstd * g[t + 256] + beta[t + 256] + (post ? post[row + t + 256] : 0.0f);
}

// Masked mean pool over sequence (intent head input). One block per batch b.
__global__ __launch_bounds__(256) void pool_kernel(
    const float* __restrict__ enc2, const int* __restrict__ mask,
    float* __restrict__ pooled, int L, int Dm)
{
  const int b = blockIdx.x;
  const int t = threadIdx.x;
  int cnt = 0;
  for (int l = 0; l < L; ++l) cnt += (mask[b * L + l] == 0);
  float inv = 1.0f / (float)cnt;
  for (int d = t; d < Dm; d += 256) {
    float acc = 0.0f;
    for (int l = 0; l < L; ++l)
      if (mask[b * L + l] == 0) acc += enc2[((size_t)(b * L + l)) * Dm + d];
    pooled[(size_t)b * Dm + d] = acc * inv;
  }
}

// Subtoken compaction: pack rows with subtok==1 to the front, zero the rest.
__global__ __launch_bounds__(256) void compact_kernel(
    const float* __restrict__ enc, const int* __restrict__ subtok,
    float* __restrict__ mem, int L, int Dm)
{
  const int b = blockIdx.x;
  const int t = threadIdx.x;
  __shared__ int dest[64];
  if (t == 0) {
    int c = 0;
    for (int l = 0; l < L; ++l) dest[l] = (subtok[b * L + l] == 1) ? c++ : -1;
  }
  __syncthreads();
  for (int i = t; i < L * Dm; i += 256) mem[(size_t)b * L * Dm + i] = 0.0f;
  __syncthreads();
  for (int l = 0; l < L; ++l) {
    int dl = dest[l];
    if (dl >= 0)
      for (int d = t; d < Dm; d += 256)
        mem[((size_t)(b * L + dl)) * Dm + d] = enc[((size_t)(b * L + l)) * Dm + d];
  }
}

// Embedding * sqrt(D) + pe[b]  (faithful to reference's pe[:B] batch-index bug;
// pe is [MAX_PE,1,D] so pe[b] broadcasts over the sequence dim).
__global__ __launch_bounds__(256) void embed_kernel(
    const int* __restrict__ ids, const float* __restrict__ emb,
    const float* __restrict__ pe, float* __restrict__ x, int L, int Dm)
{
  const int row = blockIdx.x;           // token index b*L + l
  const int b = row / L;
  const int id = ids[row];
  const int t = threadIdx.x;
  const float s = 22.62741699796952f;   // sqrt(512)
  for (int d = t; d < Dm; d += 256)
    x[(size_t)row * Dm + d] = emb[(size_t)id * Dm + d] * s + pe[(size_t)b * Dm + d];
}

// Row-wise log-softmax over C=128 classes; one block (128 threads) per row.
__global__ __launch_bounds__(128) void logsoftmax_kernel(
    const float* __restrict__ logits, float* __restrict__ out, int C)
{
  const int row = blockIdx.x;
  const int t = threadIdx.x;
  __shared__ float red[128];
  float v = logits[(size_t)row * C + t];
  red[t] = v;
  __syncthreads();
  for (int s = 64; s > 0; s >>= 1) { if (t < s) red[t] = fmaxf(red[t], red[t + s]); __syncthreads(); }
  float mx = red[0];
  __syncthreads();
  red[t] = expf(v - mx);
  __syncthreads();
  for (int s = 64; s > 0; s >>= 1) { if (t < s) red[t] += red[t + s]; __syncthreads(); }
  float lse = mx + logf(red[0]);
  out[(size_t)row * C + t] = v - lse;
}

// ---------------------------------------------------------------------------
extern "C" void kernel_launch(void* const* d_in, const int* in_sizes, int n_in,
                              void* d_out, int out_size, void* d_ws, size_t ws_size,
                              hipStream_t stream)
{
  (void)in_sizes; (void)n_in; (void)out_size; (void)ws_size;

  const int B = 512, L = 60, D = 512, FF = 300, SLOT = 128, INTENT = 26;
  const int N = B * L;                    // 30720 tokens
  const int NH_SELF = 8, NH_DEC = 2;

  // ---- inputs (setup_inputs() dict order; params flattened in insertion order)
  const int*   input    = (const int*)  d_in[0];
  const float* enc      = (const float*)d_in[1];
  const int*   encmask  = (const int*)  d_in[2];   // True(!=0) = pad
  const int*   subtok   = (const int*)  d_in[3];
  const float* emb      = (const float*)d_in[4];
  const float* e_qkv_w  = (const float*)d_in[5];
  const float* e_qkv_b  = (const float*)d_in[6];
  const float* e_out_w  = (const float*)d_in[7];
  const float* e_out_b  = (const float*)d_in[8];
  const float* ln0_g    = (const float*)d_in[9];
  const float* ln0_b    = (const float*)d_in[10];
  const float* intent_w = (const float*)d_in[11];
  const float* intent_b = (const float*)d_in[12];
  const float* slot_w   = (const float*)d_in[13];
  const float* slot_b   = (const float*)d_in[14];
  const float* pe       = (const float*)d_in[15];

  // ---- workspace layout (floats)
  float* ws   = (float*)d_ws;
  float* qkv  = ws;                                   // [N, 3D]  (aliased: ff hidden, slot logits)
  float* t1   = ws + (size_t)N * 3 * D;               // [N, D]
  float* t2   = t1 + (size_t)N * D;                   // [N, D]
  float* x    = t2 + (size_t)N * D;                   // [N, D]   decoder state
  float* mem  = x  + (size_t)N * D;                   // [N, D]   compacted memory
  float* pool = mem + (size_t)N * D;                  // [B, D]
  float* ffh  = qkv;                                  // [N, FF]  alias
  float* logits = qkv;                                // [N, SLOT] alias

  float* out_slot   = (float*)d_out;                  // [N, SLOT]
  float* out_intent = out_slot + (size_t)N * SLOT;    // [B, INTENT]

  auto gemm = [&](const float* A, const float* W, const float* bias, float* C,
                  int M, int Nc, int K, int relu) {
    dim3 grid((Nc + 63) / 64, M / 128);
    gemm_wmma_f32<<<grid, 256, 0, stream>>>(A, W, bias, C, M, Nc, K, relu);
  };

  // ================= encoder side =================
  // context = MHA(enc, enc, enc) with key-padding mask, 8 heads
  gemm(enc, e_qkv_w, e_qkv_b, qkv, N, 3 * D, D, 0);
  attn_kernel<<<B * NH_SELF * L, 64, 0, stream>>>(qkv, t1, encmask,
                                                  B, NH_SELF, L, D / NH_SELF, D, 0);
  gemm(t1, e_out_w, e_out_b, t2, N, D, D, 0);
  // enc2 = LN(context) + enc   (post-residual)
  ln_kernel<<<N, 256, 0, stream>>>(t1, t2, nullptr, enc, ln0_g, ln0_b);
  // intent = mean_keep(enc2) @ intent_w.T + b
  pool_kernel<<<B, 256, 0, stream>>>(t1, encmask, pool, L, D);
  gemm(pool, intent_w, intent_b, out_intent, B, INTENT, D, 0);

  // memory = packed subtoken rows of encoder_outputs
  compact_kernel<<<B, 256, 0, stream>>>(enc, subtok, mem, L, D);

  // x = emb[input]*sqrt(D) + pe[b]
  embed_kernel<<<N, 256, 0, stream>>>(input, emb, pe, x, L, D);

  // ================= 2 decoder layers =================
  for (int li = 0; li < 2; ++li) {
    const int base = 16 + 18 * li;
    const float* sa_qkv_w = (const float*)d_in[base + 0];
    const float* sa_qkv_b = (const float*)d_in[base + 1];
    const float* sa_out_w = (const float*)d_in[base + 2];
    const float* sa_out_b = (const float*)d_in[base + 3];
    const float* ca_qkv_w = (const float*)d_in[base + 4];
    const float* ca_qkv_b = (const float*)d_in[base + 5];
    const float* ca_out_w = (const float*)d_in[base + 6];
    const float* ca_out_b = (const float*)d_in[base + 7];
    const float* ff1_w    = (const float*)d_in[base + 8];
    const float* ff1_b    = (const float*)d_in[base + 9];
    const float* ff2_w    = (const float*)d_in[base + 10];
    const float* ff2_b    = (const float*)d_in[base + 11];
    const float* ln1_g    = (const float*)d_in[base + 12];
    const float* ln1_b    = (const float*)d_in[base + 13];
    const float* ln2_g    = (const float*)d_in[base + 14];
    const float* ln2_b    = (const float*)d_in[base + 15];
    const float* ln3_g    = (const float*)d_in[base + 16];
    const float* ln3_b    = (const float*)d_in[base + 17];

    // -- causal self-attention (2 heads, hd=256)
    gemm(x, sa_qkv_w, sa_qkv_b, qkv, N, 3 * D, D, 0);
    attn_kernel<<<B * NH_DEC * L, 64, 0, stream>>>(qkv, t1, nullptr,
                                                   B, NH_DEC, L, D / NH_DEC, D, 1);
    gemm(t1, sa_out_w, sa_out_b, t2, N, D, D, 0);
    ln_kernel<<<N, 256, 0, stream>>>(x, t2, x, nullptr, ln1_g, ln1_b);

    // -- cross-attention with diagonal mask == identity attention:
    //    softmax(diag-masked scores) is one-hot(i), so CA = out_proj(V_proj(mem)).
    gemm(mem, ca_qkv_w + (size_t)2 * D * D, ca_qkv_b + 2 * D, t1, N, D, D, 0);
    gemm(t1, ca_out_w, ca_out_b, t2, N, D, D, 0);
    ln_kernel<<<N, 256, 0, stream>>>(x, t2, x, nullptr, ln2_g, ln2_b);

    // -- feed-forward
    gemm(x, ff1_w, ff1_b, ffh, N, FF, D, 1);        // ReLU fused
    gemm(ffh, ff2_w, ff2_b, t2, N, D, FF, 0);
    ln_kernel<<<N, 256, 0, stream>>>(x, t2, x, nullptr, ln3_g, ln3_b);
  }

  // ================= heads =================
  gemm(x, slot_w, slot_b, logits, N, SLOT, D, 0);
  logsoftmax_kernel<<<N, 128, 0, stream>>>(logits, out_slot, SLOT);
}